// PartialShapeEncoder_89781996356149
// MI455X (gfx1250) — compile-verified
//
#include <hip/hip_runtime.h>
#include <hip/hip_bf16.h>
#include <stdint.h>

typedef __attribute__((ext_vector_type(2))) float v2f;
typedef __attribute__((ext_vector_type(8))) float v8f;

#define NPTS 8192
#define MSEL 4096
#define KNN  200
#define LDH  132   // padded stride for 16x128 tile (132 % 64 == 4 -> conflict-free)
#define LDX  516   // padded stride for 16x512 tile (516 % 64 == 4 -> conflict-free)

__device__ __forceinline__ float lrelu(float x) { return x > 0.f ? x : 0.01f * x; }

// monotonic float<->uint mapping so float-max == uint-max (deterministic atomics)
__device__ __forceinline__ unsigned fenc(float f) {
  unsigned u = __float_as_uint(f);
  return (u & 0x80000000u) ? ~u : (u | 0x80000000u);
}
__device__ __forceinline__ float fdec(unsigned u) {
  unsigned v = (u & 0x80000000u) ? (u & 0x7fffffffu) : ~u;
  return __uint_as_float(v);
}

// ---- CDNA5 f32 WMMA 16x16x4: exact-f32 matrix pipe -------------------------
__device__ __forceinline__ v8f wmma4(v2f a, v2f b, v8f c) {
  return __builtin_amdgcn_wmma_f32_16x16x4_f32(false, a, false, b, (short)0, c,
                                               false, false);
}
// A 16x4 layout: lane<16 -> row=lane, holds K=k0,k0+1 ; lane>=16 -> K=k0+2,k0+3
__device__ __forceinline__ v2f load_a_lds(const float* s, int ld, int k0, int lane) {
  int r  = lane & 15;
  int kk = k0 + ((lane >> 4) << 1);
  const float* p = s + r * ld + kk;
  v2f a; a.x = p[0]; a.y = p[1];
  return a;
}
// B 4x16 layout (mirror of A): lane<16 -> col=lane, K=k0,k0+1 ; lane>=16 -> K=k0+2,k0+3
__device__ __forceinline__ v2f load_b_glb(const float* __restrict__ W, int Nd,
                                          int k0, int n0, int lane) {
  int cn = n0 + (lane & 15);
  int kk = k0 + ((lane >> 4) << 1);
  v2f b; b.x = W[(size_t)kk * Nd + cn]; b.y = W[(size_t)(kk + 1) * Nd + cn];
  return b;
}

__device__ __forceinline__ unsigned long long shfl_xor_u64(unsigned long long v, int m) {
  unsigned lo = (unsigned)v, hi = (unsigned)(v >> 32);
  lo = __shfl_xor(lo, m, 32);
  hi = __shfl_xor(hi, m, 32);
  return ((unsigned long long)hi << 32) | lo;
}

// ---- 1) Farthest point sampling: one WG per batch, points in VGPRs ---------
__global__ void __launch_bounds__(1024) fps_kernel(const float* __restrict__ x,
                                                   float* __restrict__ sel) {
  int b = blockIdx.x;
  int tid = threadIdx.x, lane = tid & 31, wave = tid >> 5;
  const float* xb = x + (size_t)b * NPTS * 3;
  float px[8], py[8], pz[8], mind[8];
  float qx = xb[0], qy = xb[1], qz = xb[2];
  for (int j = 0; j < 8; ++j) {
    int idx = tid + j * 1024;
    px[j] = xb[idx * 3 + 0]; py[j] = xb[idx * 3 + 1]; pz[j] = xb[idx * 3 + 2];
    float dx = px[j] - qx, dy = py[j] - qy, dz = pz[j] - qz;
    mind[j] = dx * dx + dy * dy + dz * dz;
  }
  __shared__ float swv[32];
  __shared__ int   swi[32];
  __shared__ float spt[3];
  float* selb = sel + (size_t)b * MSEL * 3;
  if (tid == 0) { selb[0] = qx; selb[1] = qy; selb[2] = qz; }
  for (int s = 1; s < MSEL; ++s) {
    float bv = -1.f; int bi = 0;
    for (int j = 0; j < 8; ++j)
      if (mind[j] > bv) { bv = mind[j]; bi = tid + j * 1024; }
    for (int off = 16; off > 0; off >>= 1) {
      float ov = __shfl_xor(bv, off, 32);
      int   oi = __shfl_xor(bi, off, 32);
      if (ov > bv || (ov == bv && oi < bi)) { bv = ov; bi = oi; }
    }
    if (lane == 0) { swv[wave] = bv; swi[wave] = bi; }
    __syncthreads();
    if (tid == 0) {
      float gv = swv[0]; int gi = swi[0];
      for (int w = 1; w < 32; ++w) {
        float ov = swv[w]; int oi = swi[w];
        if (ov > gv || (ov == gv && oi < gi)) { gv = ov; gi = oi; }
      }
      const float* pw = xb + (size_t)gi * 3;
      spt[0] = pw[0]; spt[1] = pw[1]; spt[2] = pw[2];
      selb[s * 3 + 0] = pw[0]; selb[s * 3 + 1] = pw[1]; selb[s * 3 + 2] = pw[2];
    }
    __syncthreads();
    float wx = spt[0], wy = spt[1], wz = spt[2];
    for (int j = 0; j < 8; ++j) {
      float dx = px[j] - wx, dy = py[j] - wy, dz = pz[j] - wz;
      mind[j] = fminf(mind[j], dx * dx + dy * dy + dz * dz);
    }
    __syncthreads();
  }
}

// ---- 2) Pass 1: h = lrelu(lin1(lrelu(lin0(sel)))), pool-max via atomics ----
__global__ void __launch_bounds__(256) enc_pool_kernel(
    const float* __restrict__ sel,
    const float* __restrict__ W0, const float* __restrict__ b0,
    const float* __restrict__ W1, const float* __restrict__ b1,
    unsigned* __restrict__ pooled_u) {
  int b = blockIdx.x >> 8, tile = blockIdx.x & 255, row0 = tile * 16;
  int tid = threadIdx.x, lane = tid & 31, wave = tid >> 5;
  __shared__ float h0s[16 * LDH];
  const float* selp = sel + ((size_t)b * MSEL + row0) * 3;
  for (int o = tid; o < 16 * 128; o += 256) {
    int r = o >> 7, c = o & 127;
    const float* p = selp + r * 3;
    float s = b0[c] + p[0] * W0[c] + p[1] * W0[128 + c] + p[2] * W0[256 + c];
    h0s[r * LDH + c] = lrelu(s);
  }
  __syncthreads();
  v8f zero = {0.f, 0.f, 0.f, 0.f, 0.f, 0.f, 0.f, 0.f};
  for (int t = wave * 2; t < wave * 2 + 2; ++t) {
    int n0 = t * 16;
    v8f acc = zero;
    for (int k0 = 0; k0 < 128; k0 += 4) {
      v2f a  = load_a_lds(h0s, LDH, k0, lane);
      v2f bb = load_b_glb(W1, 256, k0, n0, lane);
      acc = wmma4(a, bb, acc);
    }
    int col = n0 + (lane & 15);
    float bias = b1[col];
    float m = -3.4e38f;
    for (int rr = 0; rr < 8; ++rr) m = fmaxf(m, lrelu(acc[rr] + bias));
    m = fmaxf(m, __shfl_xor(m, 16, 32));
    if (lane < 16) atomicMax(&pooled_u[b * 256 + col], fenc(m));
  }
}

// ---- 3) Pass 2: fused concat -> lin2 -> lin3 -> row-max into g -------------
__global__ void __launch_bounds__(256) enc_big_kernel(
    const float* __restrict__ sel,
    const float* __restrict__ W0, const float* __restrict__ b0,
    const float* __restrict__ W1, const float* __restrict__ b1,
    const float* __restrict__ W2, const float* __restrict__ b2,
    const float* __restrict__ W3, const float* __restrict__ b3,
    const unsigned* __restrict__ pooled_u,
    unsigned* __restrict__ g_u) {
  int b = blockIdx.x >> 8, tile = blockIdx.x & 255, row0 = tile * 16;
  int tid = threadIdx.x, lane = tid & 31, wave = tid >> 5;
  __shared__ float h0s[16 * LDH];
  __shared__ float Xs[16 * LDX];
  const float* selp = sel + ((size_t)b * MSEL + row0) * 3;
  v8f zero = {0.f, 0.f, 0.f, 0.f, 0.f, 0.f, 0.f, 0.f};
  // recompute h0 (cheap) + fill pooled half of the concat tile
  for (int o = tid; o < 16 * 128; o += 256) {
    int r = o >> 7, c = o & 127;
    const float* p = selp + r * 3;
    float s = b0[c] + p[0] * W0[c] + p[1] * W0[128 + c] + p[2] * W0[256 + c];
    h0s[r * LDH + c] = lrelu(s);
  }
  for (int o = tid; o < 16 * 256; o += 256) {
    int r = o >> 8, c = o & 255;
    Xs[r * LDX + c] = fdec(pooled_u[b * 256 + c]);
  }
  __syncthreads();
  // h1 tile -> Xs[:,256:512] via WMMA
  for (int t = wave * 2; t < wave * 2 + 2; ++t) {
    int n0 = t * 16;
    v8f acc = zero;
    for (int k0 = 0; k0 < 128; k0 += 4) {
      v2f a  = load_a_lds(h0s, LDH, k0, lane);
      v2f bb = load_b_glb(W1, 256, k0, n0, lane);
      acc = wmma4(a, bb, acc);
    }
    int col = n0 + (lane & 15);
    float bias = b1[col];
    for (int rr = 0; rr < 8; ++rr) {
      int row = rr + ((lane >> 4) << 3);
      Xs[row * LDX + 256 + col] = lrelu(acc[rr] + bias);
    }
  }
  __syncthreads();
  // GEMM1: T = lrelu(Xs[16x512] * W2[512x512] + b2) -> regs, then overwrite Xs
  v8f accT[4];
  for (int t = 0; t < 4; ++t) accT[t] = zero;
  int tb = wave * 4;
  for (int k0 = 0; k0 < 512; k0 += 4) {
    v2f a = load_a_lds(Xs, LDX, k0, lane);
    for (int t = 0; t < 4; ++t) {
      v2f bb = load_b_glb(W2, 512, k0, (tb + t) * 16, lane);
      accT[t] = wmma4(a, bb, accT[t]);
    }
  }
  __syncthreads();
  for (int t = 0; t < 4; ++t) {
    int col = (tb + t) * 16 + (lane & 15);
    float bias = b2[col];
    for (int rr = 0; rr < 8; ++rr) {
      int row = rr + ((lane >> 4) << 3);
      Xs[row * LDX + col] = lrelu(accT[t][rr] + bias);
    }
  }
  __syncthreads();
  // GEMM2: U = lrelu(T * W3[512x1024] + b3); max over 16 rows -> atomic into g
  v8f accU[8];
  for (int t = 0; t < 8; ++t) accU[t] = zero;
  int tb2 = wave * 8;
  for (int k0 = 0; k0 < 512; k0 += 4) {
    v2f a = load_a_lds(Xs, LDX, k0, lane);
    for (int t = 0; t < 8; ++t) {
      v2f bb = load_b_glb(W3, 1024, k0, (tb2 + t) * 16, lane);
      accU[t] = wmma4(a, bb, accU[t]);
    }
  }
  for (int t = 0; t < 8; ++t) {
    int col = (tb2 + t) * 16 + (lane & 15);
    float bias = b3[col];
    float m = -3.4e38f;
    for (int rr = 0; rr < 8; ++rr) m = fmaxf(m, lrelu(accU[t][rr] + bias));
    m = fmaxf(m, __shfl_xor(m, 16, 32));
    if (lane < 16) atomicMax(&g_u[b * 1024 + col], fenc(m));
  }
}

// ---- helpers: decode g, generic small dense layer --------------------------
__global__ void decode_kernel(const unsigned* __restrict__ u, float* __restrict__ f, int n) {
  int i = blockIdx.x * blockDim.x + threadIdx.x;
  if (i < n) f[i] = fdec(u[i]);
}

__global__ void lin_kernel(const float* __restrict__ X, const float* __restrict__ W,
                           const float* __restrict__ bias, float* __restrict__ Y,
                           float* __restrict__ Y2, int Mr, int K, int Nd, int doRelu) {
  int o = blockIdx.x * blockDim.x + threadIdx.x;
  if (o >= Mr * Nd) return;
  int r = o / Nd, c = o - r * Nd;
  float s = bias[c];
  const float* xr = X + (size_t)r * K;
  for (int k = 0; k < K; ++k) s = fmaf(xr[k], W[(size_t)k * Nd + c], s);
  if (doRelu) s = lrelu(s);
  Y[o] = s;
  if (Y2) Y2[o] = s;
}

// ---- 4) top-200 KNN per (batch, skeleton-pt): packed u64 keys in VGPRs -----
__global__ void __launch_bounds__(256) knn_kernel(const float* __restrict__ sel,
                                                  const float* __restrict__ rst1,
                                                  float* __restrict__ grp) {
  int pair = blockIdx.x;          // b*8 + q
  int b = pair >> 3;
  int tid = threadIdx.x, lane = tid & 31, wave = tid >> 5;
  const float* selb = sel + (size_t)b * MSEL * 3;
  float qx = rst1[pair * 3 + 0], qy = rst1[pair * 3 + 1], qz = rst1[pair * 3 + 2];
  unsigned long long key[16];
  for (int j = 0; j < 16; ++j) {
    int idx = tid + j * 256;
    float dx = selb[idx * 3 + 0] - qx, dy = selb[idx * 3 + 1] - qy,
          dz = selb[idx * 3 + 2] - qz;
    float d = dx * dx + dy * dy + dz * dz;        // >=0 -> IEEE bits monotonic
    key[j] = ((unsigned long long)__float_as_uint(d) << 32) | (unsigned)idx;
  }
  __shared__ unsigned long long swk[8];
  __shared__ unsigned long long swin;
  float* grpp = grp + (size_t)pair * (KNN * 3);
  for (int r = 0; r < KNN; ++r) {
    unsigned long long mk = ~0ull;
    for (int j = 0; j < 16; ++j) mk = (key[j] < mk) ? key[j] : mk;
    for (int off = 16; off > 0; off >>= 1) {
      unsigned long long ok = shfl_xor_u64(mk, off);
      mk = (ok < mk) ? ok : mk;
    }
    if (lane == 0) swk[wave] = mk;
    __syncthreads();
    if (tid == 0) {
      unsigned long long g = swk[0];
      for (int w = 1; w < 8; ++w) g = (swk[w] < g) ? swk[w] : g;
      swin = g;
      int widx = (int)(unsigned)(g & 0xffffffffull);
      grpp[r * 3 + 0] = selb[widx * 3 + 0];
      grpp[r * 3 + 1] = selb[widx * 3 + 1];
      grpp[r * 3 + 2] = selb[widx * 3 + 2];
    }
    __syncthreads();
    unsigned long long wmin = swin;
    for (int j = 0; j < 16; ++j) if (key[j] == wmin) key[j] = ~0ull;
    __syncthreads();
  }
}

extern "C" void kernel_launch(void* const* d_in, const int* in_sizes, int n_in,
                              void* d_out, int out_size, void* d_ws, size_t ws_size,
                              hipStream_t stream) {
  (void)in_sizes; (void)n_in; (void)out_size; (void)ws_size;
  const float* x  = (const float*)d_in[0];
  const float* W0 = (const float*)d_in[3];  const float* b0 = (const float*)d_in[4];
  const float* W1 = (const float*)d_in[5];  const float* b1 = (const float*)d_in[6];
  const float* W2 = (const float*)d_in[7];  const float* b2 = (const float*)d_in[8];
  const float* W3 = (const float*)d_in[9];  const float* b3 = (const float*)d_in[10];
  const float* W4 = (const float*)d_in[11]; const float* b4 = (const float*)d_in[12];
  const float* W5 = (const float*)d_in[13]; const float* b5 = (const float*)d_in[14];
  const float* W6 = (const float*)d_in[15]; const float* b6 = (const float*)d_in[16];
  const float* W7 = (const float*)d_in[17]; const float* b7 = (const float*)d_in[18];
  const float* W8 = (const float*)d_in[19]; const float* b8 = (const float*)d_in[20];
  const float* W9 = (const float*)d_in[21]; const float* b9 = (const float*)d_in[22];
  const float* W10= (const float*)d_in[23]; const float* b10= (const float*)d_in[24];
  const float* W11= (const float*)d_in[25]; const float* b11= (const float*)d_in[26];

  char* w = (char*)d_ws;
  float*    sel      = (float*)(w + 0);          // 16*4096*3
  unsigned* pooled_u = (unsigned*)(w + 786432);  // 16*256
  unsigned* g_u      = (unsigned*)(w + 802816);  // 16*1024
  float*    g_f      = (float*)(w + 868352);     // 16*1024
  float*    a4       = (float*)(w + 933888);     // 16*512
  float*    a5       = (float*)(w + 966656);     // 16*128
  float*    rst1     = (float*)(w + 974848);     // 128*3
  float*    grp      = (float*)(w + 976384);     // 128*600
  float*    h2a      = (float*)(w + 1283584);    // 128*256
  float*    h2b      = (float*)(w + 1414656);    // 128*256
  float*    h2c      = (float*)(w + 1545728);    // 128*64
  float*    h2d      = (float*)(w + 1578496);    // 16*512
  float*    outp     = (float*)d_out;            // [0,384)=rst1, [384,2432)=out

  hipMemsetAsync(pooled_u, 0, 16 * 256 * 4, stream);
  hipMemsetAsync(g_u,      0, 16 * 1024 * 4, stream);

  fps_kernel<<<16, 1024, 0, stream>>>(x, sel);
  enc_pool_kernel<<<4096, 256, 0, stream>>>(sel, W0, b0, W1, b1, pooled_u);
  enc_big_kernel<<<4096, 256, 0, stream>>>(sel, W0, b0, W1, b1, W2, b2, W3, b3,
                                           pooled_u, g_u);
  decode_kernel<<<64, 256, 0, stream>>>(g_u, g_f, 16 * 1024);
  lin_kernel<<<32, 256, 0, stream>>>(g_f, W4, b4, a4, nullptr, 16, 1024, 512, 1);
  lin_kernel<<<8, 256, 0, stream>>>(a4, W5, b5, a5, nullptr, 16, 512, 128, 1);
  lin_kernel<<<2, 256, 0, stream>>>(a5, W6, b6, rst1, outp, 16, 128, 24, 0);
  knn_kernel<<<128, 256, 0, stream>>>(sel, rst1, grp);
  lin_kernel<<<128, 256, 0, stream>>>(grp, W7, b7, h2a, nullptr, 128, 600, 256, 1);
  lin_kernel<<<128, 256, 0, stream>>>(h2a, W8, b8, h2b, nullptr, 128, 256, 256, 1);
  lin_kernel<<<32, 256, 0, stream>>>(h2b, W9, b9, h2c, nullptr, 128, 256, 64, 1);
  lin_kernel<<<32, 256, 0, stream>>>(h2c, W10, b10, h2d, nullptr, 16, 512, 512, 1);
  lin_kernel<<<8, 256, 0, stream>>>(h2d, W11, b11, outp + 384, nullptr, 16, 512, 128, 0);
}